// HungarianMatcher_12257836663341
// MI455X (gfx1250) — compile-verified
//
#include <hip/hip_runtime.h>
#include <hip/hip_bf16.h>
#include <stdint.h>

// ---------------------------------------------------------------------------
// OTA / dynamic-k Hungarian-style matcher for MI455X (gfx1250, wave32, WMMA).
// cost_class gather is computed as an EXACT f32 GEMM against a one-hot matrix
// using V_WMMA_F32_16X16X4_F32 (one wave per 16x16 output tile, 20 k-steps).
// ---------------------------------------------------------------------------

#define ALPHA_F 0.25f
#define W_CLASS 2.0f
#define W_BBOX 5.0f
#define W_GIOU 2.0f
#define CENTER_RADIUS 2.5f

#define Bn 32
#define Qn 2000
#define Gn 64
#define Cn 80

typedef __attribute__((ext_vector_type(2))) float v2f;
typedef __attribute__((ext_vector_type(8))) float v8f;

// ---------------------------------------------------------------------------
// Workspace layout (bytes):
//   T    : [B,Q,C] f32  class table (pos - neg)           @ 0         (20,480,000)
//   cost : [B,G,Q] f32  column-major cost                 @ 20,480,000 (16,384,000)
//   mask : [B,Q]   u64  match bitmask over gts            @ 36,864,000 (512,000)
//   keep : [B,Q]   u8   argmin_g cost (row keep gt)       @ 37,376,000 (64,000)
// ---------------------------------------------------------------------------
#define WS_T_OFF    0
#define WS_COST_OFF 20480000
#define WS_MASK_OFF 36864000
#define WS_KEEP_OFF 37376000

// -------------------------- init: zero match bitmask ------------------------
__global__ void hm_init_mask(unsigned long long* __restrict__ mask) {
  int i = blockIdx.x * blockDim.x + threadIdx.x;
  if (i < Bn * Qn) mask[i] = 0ull;
}

// ------------------- focal class table T = pos - neg ------------------------
__global__ void hm_class_table(const float* __restrict__ logits,
                               float* __restrict__ T) {
  size_t i = (size_t)blockIdx.x * blockDim.x + threadIdx.x;
  if (i >= (size_t)Bn * Qn * Cn) return;
  float p = logits[i];
  float negv = (1.0f - ALPHA_F) * p * p * (-logf(1.0f - p + 1e-8f));
  float posv = ALPHA_F * (1.0f - p) * (1.0f - p) * (-logf(p + 1e-8f));
  T[i] = posv - negv;
}

// -------- geometric cost: L1 bbox + GIoU + center-prior penalties -----------
__global__ void hm_geom_cost(const float* __restrict__ pred_boxes,
                             const float* __restrict__ gt_boxes,
                             const float* __restrict__ img_sz,
                             const float* __restrict__ img_sz_tgt,
                             float* __restrict__ cost) {
  __shared__ float sGT[Gn * 4];
  __shared__ float sTS[Gn * 4];
  const int b = blockIdx.y;
  for (int i = threadIdx.x; i < Gn * 4; i += blockDim.x) {
    sGT[i] = gt_boxes[(size_t)b * Gn * 4 + i];
    sTS[i] = img_sz_tgt[(size_t)b * Gn * 4 + i];
  }
  __syncthreads();
  const int q = blockIdx.x * blockDim.x + threadIdx.x;
  if (q >= Qn) return;

  const float4 pb = *(const float4*)(pred_boxes + ((size_t)b * Qn + q) * 4);
  const float ax1 = pb.x, ay1 = pb.y, ax2 = pb.z, ay2 = pb.w;
  const float cx = 0.5f * (ax1 + ax2), cy = 0.5f * (ay1 + ay2);
  const float areaA = (ax2 - ax1) * (ay2 - ay1);
  const float s0 = img_sz[b * 4 + 0], s1 = img_sz[b * 4 + 1];
  const float s2 = img_sz[b * 4 + 2], s3 = img_sz[b * 4 + 3];
  const float nx1 = ax1 / s0, ny1 = ay1 / s1, nx2 = ax2 / s2, ny2 = ay2 / s3;

  // pass 1: fg mask (any-in-box OR any-in-center over gts)
  bool anyB = false, anyC = false;
  for (int g = 0; g < Gn; ++g) {
    const float gx1 = sGT[g * 4 + 0], gy1 = sGT[g * 4 + 1];
    const float gx2 = sGT[g * 4 + 2], gy2 = sGT[g * 4 + 3];
    const bool inB = (cx > gx1) && (cx < gx2) && (cy > gy1) && (cy < gy2);
    const float gcx = 0.5f * (gx1 + gx2), gcy = 0.5f * (gy1 + gy2);
    const float gw = gx2 - gx1, gh = gy2 - gy1;
    const bool inC = (cx > gcx - CENTER_RADIUS * gw) && (cx < gcx + CENTER_RADIUS * gw) &&
                     (cy > gcy - CENTER_RADIUS * gh) && (cy < gcy + CENTER_RADIUS * gh);
    anyB |= inB; anyC |= inC;
  }
  const float fgAdd = (anyB || anyC) ? 0.0f : 1e4f;

  // pass 2: per-pair cost (class term added later by WMMA kernel)
  for (int g = 0; g < Gn; ++g) {
    const float gx1 = sGT[g * 4 + 0], gy1 = sGT[g * 4 + 1];
    const float gx2 = sGT[g * 4 + 2], gy2 = sGT[g * 4 + 3];
    const bool inB = (cx > gx1) && (cx < gx2) && (cy > gy1) && (cy < gy2);
    const float gcx = 0.5f * (gx1 + gx2), gcy = 0.5f * (gy1 + gy2);
    const float gw = gx2 - gx1, gh = gy2 - gy1;
    const bool inC = (cx > gcx - CENTER_RADIUS * gw) && (cx < gcx + CENTER_RADIUS * gw) &&
                     (cy > gcy - CENTER_RADIUS * gh) && (cy < gcy + CENTER_RADIUS * gh);
    const bool inBC = inB && inC;

    const float areaG = (gx2 - gx1) * (gy2 - gy1);
    const float iw = fmaxf(fminf(ax2, gx2) - fmaxf(ax1, gx1), 0.0f);
    const float ih = fmaxf(fminf(ay2, gy2) - fmaxf(ay1, gy1), 0.0f);
    const float inter = iw * ih;
    const float uni = areaA + areaG - inter;
    const float iou = inter / uni;
    const float cw = fmaxf(fmaxf(ax2, gx2) - fminf(ax1, gx1), 0.0f);
    const float ch = fmaxf(fmaxf(ay2, gy2) - fminf(ay1, gy1), 0.0f);
    const float areaC = cw * ch;
    const float giou = iou - (areaC - uni) / areaC;

    const float tx1 = gx1 / sTS[g * 4 + 0], ty1 = gy1 / sTS[g * 4 + 1];
    const float tx2 = gx2 / sTS[g * 4 + 2], ty2 = gy2 / sTS[g * 4 + 3];
    const float cbbox = fabsf(nx1 - tx1) + fabsf(ny1 - ty1) +
                        fabsf(nx2 - tx2) + fabsf(ny2 - ty2);

    const float cval = W_BBOX * cbbox - W_GIOU * giou +
                       (inBC ? 0.0f : 100.0f) + fgAdd;
    cost[((size_t)b * Gn + g) * Qn + q] = cval;  // column-major [b][g][q]
  }
}

// ------------- WMMA: cost += W_CLASS * (T[Q,C] x onehot[C,G]) ---------------
// One wave per 16x16 (q,g) tile; K accumulated over C=80 in 20 WMMA x4 steps.
// A (16x4 f32) layout per ISA: lanes 0-15 M=0..15; VGPR0=K0|K2, VGPR1=K1|K3.
// B (4x16 f32) mirrored; C/D: VGPR r -> M=r (lanes 0-15) / M=r+8 (lanes 16-31).
__global__ void hm_wmma_class(const float* __restrict__ T,
                              const int* __restrict__ gt_classes,
                              float* __restrict__ cost) {
  const int qt = blockIdx.x;      // 0..124
  const int gt = blockIdx.y;      // 0..3
  const int b  = blockIdx.z;      // 0..31
  const int lane = threadIdx.x;   // 0..31 (wave32)
  const int n  = lane & 15;
  const int kh = lane >> 4;       // 0 or 1
  const int q0 = qt * 16, g0 = gt * 16;

  const int cls = gt_classes[b * Gn + g0 + n];
  const float* Trow = T + ((size_t)b * Qn + (q0 + n)) * Cn;  // A row M = lane&15

  v8f acc = {};
  for (int c0 = 0; c0 < Cn; c0 += 4) {
    const int k = c0 + kh * 2;
    v2f a;  a.x  = Trow[k];              a.y  = Trow[k + 1];
    v2f bm; bm.x = (cls == k) ? 1.0f : 0.0f;
            bm.y = (cls == k + 1) ? 1.0f : 0.0f;
    acc = __builtin_amdgcn_wmma_f32_16x16x4_f32(
        /*neg_a=*/false, a, /*neg_b=*/false, bm,
        /*c_mod=*/(short)0, acc, /*reuse_a=*/false, /*reuse_b=*/false);
  }

#pragma unroll
  for (int r = 0; r < 8; ++r) {
    const int q = q0 + r + 8 * kh;
    const int g = g0 + n;
    cost[((size_t)b * Gn + g) * Qn + q] += W_CLASS * acc[r];
  }
}

// --------------- per-row keep gt = argmin_g cost (tie -> first) -------------
__global__ void hm_keep(const float* __restrict__ cost,
                        unsigned char* __restrict__ keep) {
  const int i = blockIdx.x * blockDim.x + threadIdx.x;
  if (i >= Bn * Qn) return;
  const int b = i / Qn, q = i % Qn;
  float bv = cost[((size_t)b * Gn + 0) * Qn + q];
  int bg = 0;
  for (int g = 1; g < Gn; ++g) {
    const float v = cost[((size_t)b * Gn + g) * Qn + q];
    if (v < bv) { bv = v; bg = g; }
  }
  keep[i] = (unsigned char)bg;
}

// -------- per-column top-5: dynamic_k from ious, k cheapest -> mask bits ----
__device__ inline void merge_desc5(float* a, const float* b) {
  float m[5]; int i = 0, j = 0;
#pragma unroll
  for (int t = 0; t < 5; ++t) {
    const bool ta = (i < 5) && (j >= 5 || a[i] >= b[j]);
    m[t] = ta ? a[i++] : b[j++];
  }
#pragma unroll
  for (int t = 0; t < 5; ++t) a[t] = m[t];
}

__device__ inline void merge_asc5(float* ac, int* ai, const float* bc, const int* bi) {
  float mc[5]; int mi[5]; int i = 0, j = 0;
#pragma unroll
  for (int t = 0; t < 5; ++t) {
    const bool ta = (i < 5) && (j >= 5 || (ac[i] < bc[j]) ||
                                (ac[i] == bc[j] && ai[i] <= bi[j]));
    if (ta) { mc[t] = ac[i]; mi[t] = ai[i]; ++i; }
    else    { mc[t] = bc[j]; mi[t] = bi[j]; ++j; }
  }
#pragma unroll
  for (int t = 0; t < 5; ++t) { ac[t] = mc[t]; ai[t] = mi[t]; }
}

__global__ void hm_topk(const float* __restrict__ pred_boxes,
                        const float* __restrict__ gt_boxes,
                        const float* __restrict__ cost,
                        unsigned long long* __restrict__ mask) {
  const int col = blockIdx.x;               // b*Gn + g
  const int b = col / Gn, g = col % Gn;
  const int lane = threadIdx.x;             // wave32

  const float4 gb = *(const float4*)(gt_boxes + (size_t)col * 4);
  const float gx1 = gb.x, gy1 = gb.y, gx2 = gb.z, gy2 = gb.w;
  const float areaG = (gx2 - gx1) * (gy2 - gy1);
  const float* colc = cost + (size_t)col * Qn;

  float t5[5] = {-1e30f, -1e30f, -1e30f, -1e30f, -1e30f};        // ious, desc
  float c5[5] = {1e30f, 1e30f, 1e30f, 1e30f, 1e30f};             // cost, asc
  int   i5[5] = {0x7fffffff, 0x7fffffff, 0x7fffffff, 0x7fffffff, 0x7fffffff};

  for (int q = lane; q < Qn; q += 32) {
    const float4 pb = *(const float4*)(pred_boxes + ((size_t)b * Qn + q) * 4);
    const float areaA = (pb.z - pb.x) * (pb.w - pb.y);
    const float iw = fmaxf(fminf(pb.z, gx2) - fmaxf(pb.x, gx1), 0.0f);
    const float ih = fmaxf(fminf(pb.w, gy2) - fmaxf(pb.y, gy1), 0.0f);
    const float inter = iw * ih;
    const float iou = inter / (areaA + areaG - inter);
    if (iou > t5[4]) {
      t5[4] = iou;
#pragma unroll
      for (int t = 4; t > 0; --t)
        if (t5[t] > t5[t - 1]) { const float tmp = t5[t]; t5[t] = t5[t - 1]; t5[t - 1] = tmp; }
    }
    const float c = colc[q];
    if ((c < c5[4]) || (c == c5[4] && q < i5[4])) {
      c5[4] = c; i5[4] = q;
#pragma unroll
      for (int t = 4; t > 0; --t) {
        if ((c5[t] < c5[t - 1]) || (c5[t] == c5[t - 1] && i5[t] < i5[t - 1])) {
          const float tc = c5[t]; c5[t] = c5[t - 1]; c5[t - 1] = tc;
          const int ti = i5[t]; i5[t] = i5[t - 1]; i5[t - 1] = ti;
        }
      }
    }
  }

  for (int off = 16; off; off >>= 1) {
    float pt[5], pc[5]; int pi[5];
#pragma unroll
    for (int t = 0; t < 5; ++t) {
      pt[t] = __shfl_down(t5[t], off);
      pc[t] = __shfl_down(c5[t], off);
      pi[t] = __shfl_down(i5[t], off);
    }
    merge_desc5(t5, pt);
    merge_asc5(c5, i5, pc, pi);
  }

  if (lane == 0) {
    const float s = t5[0] + t5[1] + t5[2] + t5[3] + t5[4];
    int k = (int)s;
    if (k < 1) k = 1;
    if (k > 5) k = 5;
    for (int j = 0; j < k; ++j)
      atomicOr(&mask[(size_t)b * Qn + i5[j]], 1ull << g);
  }
}

// ------------- per-batch fix-up loop + dedup + final outputs ----------------
__global__ void hm_fixup(const float* __restrict__ cost,
                         const unsigned long long* __restrict__ mask,
                         const unsigned char* __restrict__ keep,
                         float* __restrict__ out_matching,
                         float* __restrict__ out_ids) {
  __shared__ unsigned long long sMq[Qn];
  __shared__ float sPen[Qn];
  __shared__ unsigned char sKg[Qn];
  __shared__ unsigned long long sOr[8];
  __shared__ int sQ[Gn];
  __shared__ unsigned long long sUn;

  const int b = blockIdx.x;
  const int tid = threadIdx.x;          // 256 threads = 8 waves
  const int lane = tid & 31, wid = tid >> 5;

  for (int q = tid; q < Qn; q += 256) {
    sMq[q] = mask[(size_t)b * Qn + q];
    sPen[q] = 0.0f;
    sKg[q] = keep[(size_t)b * Qn + q];
  }
  __syncthreads();

  // initial dedup: rows matched to >1 gt keep only their min-cost gt
  for (int q = tid; q < Qn; q += 256)
    if (__popcll(sMq[q]) > 1) sMq[q] = 1ull << sKg[q];
  __syncthreads();

  for (int it = 0; it < Gn; ++it) {
    // matched-columns OR reduction
    unsigned long long loc = 0ull;
    for (int q = tid; q < Qn; q += 256) loc |= sMq[q];
    for (int off = 16; off; off >>= 1) loc |= __shfl_down(loc, off);
    if (lane == 0) sOr[wid] = loc;
    __syncthreads();
    if (tid == 0) {
      unsigned long long f = 0ull;
      for (int i = 0; i < 8; ++i) f |= sOr[i];
      sUn = ~f;
    }
    __syncthreads();
    const unsigned long long um = sUn;
    if (um == 0ull) break;

    // penalize currently matched rows (accumulates across iterations)
    for (int q = tid; q < Qn; q += 256)
      if (sMq[q]) sPen[q] += 1e5f;
    __syncthreads();

    // column argmin for each unmatched gt (one wave per column)
    for (int g = wid; g < Gn; g += 8) {
      int qbest = -1;
      if ((um >> g) & 1ull) {
        const float* col = cost + ((size_t)b * Gn + g) * Qn;
        float bv = 3.0e38f; int bq = 0x7fffffff;
        for (int q = lane; q < Qn; q += 32) {
          const float v = col[q] + sPen[q];
          if (v < bv || (v == bv && q < bq)) { bv = v; bq = q; }
        }
        for (int off = 16; off; off >>= 1) {
          const float ov = __shfl_down(bv, off);
          const int oq = __shfl_down(bq, off);
          if (ov < bv || (ov == bv && oq < bq)) { bv = ov; bq = oq; }
        }
        qbest = bq;
      }
      if (lane == 0) sQ[g] = qbest;
    }
    __syncthreads();
    if (tid < Gn) {
      const int qs = sQ[tid];
      if (qs >= 0) atomicOr(&sMq[qs], 1ull << tid);
    }
    __syncthreads();
    // dedup (row keep-gt is invariant to row-constant penalties)
    for (int q = tid; q < Qn; q += 256)
      if (__popcll(sMq[q]) > 1) sMq[q] = 1ull << sKg[q];
    __syncthreads();
  }

  // matched_query_id[g] = argmin_q masked final cost (all-inf column -> 0)
  for (int g = wid; g < Gn; g += 8) {
    const float* col = cost + ((size_t)b * Gn + g) * Qn;
    float bv = __builtin_inff(); int bq = 0x7fffffff;
    for (int q = lane; q < Qn; q += 32) {
      if ((sMq[q] >> g) & 1ull) {
        const float v = col[q] + sPen[q];
        if (v < bv || (v == bv && q < bq)) { bv = v; bq = q; }
      }
    }
    for (int off = 16; off; off >>= 1) {
      const float ov = __shfl_down(bv, off);
      const int oq = __shfl_down(bq, off);
      if (ov < bv || (ov == bv && oq < bq)) { bv = ov; bq = oq; }
    }
    if (lane == 0) out_ids[b * Gn + g] = (float)((bq == 0x7fffffff) ? 0 : bq);
  }

  // matching [B,Q,G] as 0/1 floats
  for (int q = tid; q < Qn; q += 256) {
    const unsigned long long m = sMq[q];
    float* row = out_matching + ((size_t)b * Qn + q) * Gn;
    for (int g = 0; g < Gn; ++g) row[g] = (float)((m >> g) & 1ull);
  }
}

// ---------------------------------------------------------------------------
extern "C" void kernel_launch(void* const* d_in, const int* in_sizes, int n_in,
                              void* d_out, int out_size, void* d_ws, size_t ws_size,
                              hipStream_t stream) {
  (void)in_sizes; (void)n_in; (void)out_size; (void)ws_size;
  const float* pred_logits  = (const float*)d_in[0];   // [B,Q,C]
  const float* pred_boxes   = (const float*)d_in[1];   // [B,Q,4]
  const int*   gt_classes   = (const int*)d_in[2];     // [B,G]
  const float* gt_boxes     = (const float*)d_in[3];   // [B,G,4]
  const float* img_sz       = (const float*)d_in[4];   // [B,4]
  const float* img_sz_tgt   = (const float*)d_in[5];   // [B,G,4]

  char* ws = (char*)d_ws;
  float* T = (float*)(ws + WS_T_OFF);
  float* cost = (float*)(ws + WS_COST_OFF);
  unsigned long long* mask = (unsigned long long*)(ws + WS_MASK_OFF);
  unsigned char* keep = (unsigned char*)(ws + WS_KEEP_OFF);

  float* out_matching = (float*)d_out;
  float* out_ids = out_matching + (size_t)Bn * Qn * Gn;

  hm_init_mask<<<(Bn * Qn + 255) / 256, 256, 0, stream>>>(mask);
  hm_class_table<<<((size_t)Bn * Qn * Cn + 255) / 256, 256, 0, stream>>>(pred_logits, T);
  hm_geom_cost<<<dim3((Qn + 255) / 256, Bn), 256, 0, stream>>>(
      pred_boxes, gt_boxes, img_sz, img_sz_tgt, cost);
  hm_wmma_class<<<dim3(Qn / 16, Gn / 16, Bn), 32, 0, stream>>>(T, gt_classes, cost);
  hm_keep<<<(Bn * Qn + 255) / 256, 256, 0, stream>>>(cost, keep);
  hm_topk<<<Bn * Gn, 32, 0, stream>>>(pred_boxes, gt_boxes, cost, mask);
  hm_fixup<<<Bn, 256, 0, stream>>>(cost, mask, keep, out_matching, out_ids);
}